// GCNLayer_44126493999600
// MI455X (gfx1250) — compile-verified
//
#include <hip/hip_runtime.h>
#include <hip/hip_bf16.h>

typedef __attribute__((ext_vector_type(2))) float v2f;
typedef __attribute__((ext_vector_type(4))) float v4f;
typedef __attribute__((ext_vector_type(8))) float v8f;

#define GCN_B 8
#define GCN_N 512
#define GCN_ND 64
#define GCN_ED 16
#define GCN_OD 64

// ---------------------------------------------------------------------------
// K1: masked edge-feature aggregation (bandwidth-bound, ~142 MB streamed).
// One block per (b,i) row. eagg[row][e] = sum_j adj[row][j]*ef[row][j][e],
// deg[row] = sum_j adj[row][j].
// ---------------------------------------------------------------------------
__global__ __launch_bounds__(256) void gcn_agg_kernel(
    const int* __restrict__ adj,        // (B*N, N)
    const float* __restrict__ edge_feat,// (B*N, N, ED)
    float* __restrict__ eagg,           // (B*N, ED)
    float* __restrict__ degout)         // (B*N)
{
    __shared__ float sadj[GCN_N];
    __shared__ v4f   spart[256];
    __shared__ float sdeg[256];

    const int t   = threadIdx.x;
    const int row = blockIdx.x;
    const int*   arow = adj       + (size_t)row * GCN_N;
    const float* erow = edge_feat + (size_t)row * GCN_N * GCN_ED;

    // Stage adjacency row in LDS (as float) and reduce degree.
    const float a0 = (float)arow[t];
    const float a1 = (float)arow[t + 256];
    sadj[t]       = a0;
    sadj[t + 256] = a1;
    sdeg[t] = a0 + a1;
    __syncthreads();
    #pragma unroll
    for (int off = 128; off > 0; off >>= 1) {
        if (t < off) sdeg[t] += sdeg[t + off];
        __syncthreads();
    }
    const float deg = sdeg[0];
    __syncthreads();

    // Each thread: feature chunk q (4 floats of the 16), j-stripe jg, jg+64, ...
    const int q  = t & 3;
    const int jg = t >> 2;                 // 0..63
    v4f acc = (v4f){0.f, 0.f, 0.f, 0.f};
    #pragma unroll
    for (int it = 0; it < 8; ++it) {
        const int j = jg + it * 64;
        const float aj = sadj[j];
        const v4f ef = __builtin_nontemporal_load(
            (const v4f*)(erow + (size_t)j * GCN_ED + q * 4));
        acc.x = fmaf(aj, ef.x, acc.x);
        acc.y = fmaf(aj, ef.y, acc.y);
        acc.z = fmaf(aj, ef.z, acc.z);
        acc.w = fmaf(aj, ef.w, acc.w);
    }

    // Tree-reduce over the 64 j-stripes (stride-4 keeps q aligned).
    spart[t] = acc;
    __syncthreads();
    #pragma unroll
    for (int off = 128; off >= 4; off >>= 1) {
        if (t < off) {
            v4f m = spart[t];
            const v4f o = spart[t + off];
            m.x += o.x; m.y += o.y; m.z += o.z; m.w += o.w;
            spart[t] = m;
        }
        __syncthreads();
    }
    if (t < 4) {
        *(v4f*)(eagg + (size_t)row * GCN_ED + t * 4) = spart[t];
    }
    if (t == 0) degout[row] = deg;
}

// ---------------------------------------------------------------------------
// K2: fused GEMM + epilogue via V_WMMA_F32_16X16X4_F32 (full f32 precision).
// One wave (32 lanes) per 16-row tile; 4 N-tiles of 16 cover OD=64.
// out = relu( Wn@(s*h) + Ws@h + We@(r*eagg) + s*(Wn_b+We_b) + Ws_b )
// where s = deg/max(deg,1), r = 1/max(deg,1), folded per-row into the
// A fragment (A layout: lane%16 == M row; VGPR/lane-half == K).
// ---------------------------------------------------------------------------
__global__ __launch_bounds__(32) void gcn_wmma_kernel(
    const float* __restrict__ h,     // (B*N, ND)
    const float* __restrict__ eagg,  // (B*N, ED)
    const float* __restrict__ deg,   // (B*N)
    const float* __restrict__ Wn_w,  // (OD, ND)
    const float* __restrict__ Wn_b,  // (OD)
    const float* __restrict__ We_w,  // (OD, ED)
    const float* __restrict__ We_b,  // (OD)
    const float* __restrict__ Ws_w,  // (OD, ND)
    const float* __restrict__ Ws_b,  // (OD)
    float* __restrict__ out)         // (B*N, OD)
{
    const int lane    = threadIdx.x;       // 0..31, all active (EXEC all 1s)
    const int half    = lane >> 4;         // lane-half selects K pair
    const int lm      = lane & 15;         // M row (A) / N col (B, C/D)
    const int rowtile = blockIdx.x;
    const int row     = rowtile * 16 + lm;
    const int kk      = 2 * half;

    const float d  = deg[row];
    const float cd = fmaxf(d, 1.0f);
    const float s  = d / cd;
    const float r  = 1.0f / cd;

    const float* hrow = h    + (size_t)row * GCN_ND;
    const float* erow = eagg + (size_t)row * GCN_ED;

    v8f acc[4];
    #pragma unroll
    for (int nt = 0; nt < 4; ++nt) acc[nt] = (v8f){0,0,0,0,0,0,0,0};

    // Term 1: Wn @ (s*h), K = 64 (16 k-steps of 4)
    #pragma unroll
    for (int ks = 0; ks < 16; ++ks) {
        const int kb = ks * 4;
        v2f a;
        a.x = s * hrow[kb + kk];
        a.y = s * hrow[kb + kk + 1];
        #pragma unroll
        for (int nt = 0; nt < 4; ++nt) {
            const int o = nt * 16 + lm;
            v2f b;
            b.x = Wn_w[(size_t)o * GCN_ND + kb + kk];
            b.y = Wn_w[(size_t)o * GCN_ND + kb + kk + 1];
            acc[nt] = __builtin_amdgcn_wmma_f32_16x16x4_f32(
                false, a, false, b, (short)0, acc[nt], false, false);
        }
    }

    // Term 2: Ws @ h, K = 64
    #pragma unroll
    for (int ks = 0; ks < 16; ++ks) {
        const int kb = ks * 4;
        v2f a;
        a.x = hrow[kb + kk];
        a.y = hrow[kb + kk + 1];
        #pragma unroll
        for (int nt = 0; nt < 4; ++nt) {
            const int o = nt * 16 + lm;
            v2f b;
            b.x = Ws_w[(size_t)o * GCN_ND + kb + kk];
            b.y = Ws_w[(size_t)o * GCN_ND + kb + kk + 1];
            acc[nt] = __builtin_amdgcn_wmma_f32_16x16x4_f32(
                false, a, false, b, (short)0, acc[nt], false, false);
        }
    }

    // Term 3: We @ (r*eagg), K = 16 (4 k-steps)
    #pragma unroll
    for (int ks = 0; ks < 4; ++ks) {
        const int kb = ks * 4;
        v2f a;
        a.x = r * erow[kb + kk];
        a.y = r * erow[kb + kk + 1];
        #pragma unroll
        for (int nt = 0; nt < 4; ++nt) {
            const int o = nt * 16 + lm;
            v2f b;
            b.x = We_w[(size_t)o * GCN_ED + kb + kk];
            b.y = We_w[(size_t)o * GCN_ED + kb + kk + 1];
            acc[nt] = __builtin_amdgcn_wmma_f32_16x16x4_f32(
                false, a, false, b, (short)0, acc[nt], false, false);
        }
    }

    // Term 4 (bias as 2 extra K columns): A cols = [s, 1, 0, 0],
    // B rows = [Wn_b+We_b, Ws_b, 0, 0]  =>  adds s*(Wn_b+We_b)+Ws_b per row.
    {
        v2f a;
        a.x = half ? 0.0f : s;
        a.y = half ? 0.0f : 1.0f;
        #pragma unroll
        for (int nt = 0; nt < 4; ++nt) {
            const int o = nt * 16 + lm;
            const float bn = Wn_b[o] + We_b[o];
            const float bs = Ws_b[o];
            v2f b;
            b.x = half ? 0.0f : bn;
            b.y = half ? 0.0f : bs;
            acc[nt] = __builtin_amdgcn_wmma_f32_16x16x4_f32(
                false, a, false, b, (short)0, acc[nt], false, false);
        }
    }

    // Epilogue: ReLU + store. C/D layout: VGPR v -> M = v + 8*half, N = lm.
    #pragma unroll
    for (int nt = 0; nt < 4; ++nt) {
        #pragma unroll
        for (int v = 0; v < 8; ++v) {
            const int m = rowtile * 16 + v + 8 * half;
            out[(size_t)m * GCN_OD + nt * 16 + lm] = fmaxf(acc[nt][v], 0.0f);
        }
    }
}

extern "C" void kernel_launch(void* const* d_in, const int* in_sizes, int n_in,
                              void* d_out, int out_size, void* d_ws, size_t ws_size,
                              hipStream_t stream) {
    const float* h    = (const float*)d_in[0];
    const int*   adj  = (const int*)  d_in[1];
    const float* ef   = (const float*)d_in[2];
    const float* Wn_w = (const float*)d_in[3];
    const float* Wn_b = (const float*)d_in[4];
    const float* We_w = (const float*)d_in[5];
    const float* We_b = (const float*)d_in[6];
    const float* Ws_w = (const float*)d_in[7];
    const float* Ws_b = (const float*)d_in[8];
    float* out = (float*)d_out;

    const int rows = GCN_B * GCN_N;            // 4096
    float* eagg = (float*)d_ws;                // rows*ED floats
    float* degw = eagg + (size_t)rows * GCN_ED;// rows floats

    gcn_agg_kernel<<<rows, 256, 0, stream>>>(adj, ef, eagg, degw);
    gcn_wmma_kernel<<<rows / 16, 32, 0, stream>>>(
        h, eagg, degw, Wn_w, Wn_b, We_w, We_b, Ws_w, Ws_b, out);
}